// MultiHeadTemporalAttention_49581102465257
// MI455X (gfx1250) — compile-verified
//
#include <hip/hip_runtime.h>
#include <hip/hip_bf16.h>

typedef __attribute__((ext_vector_type(2))) float v2f;
typedef __attribute__((ext_vector_type(8))) float v8f;

#define D_MODEL 1024
#define N_HEADS 16
#define D_K     64
#define SEQ     2048
#define BATCH   4
#define BT      (BATCH * SEQ)        // 8192 tokens
#define LN_EPS  1e-5f
#define SPAD    (SEQ + 4)            // LDS row pitch (floats): stride % 64 banks == 4

// ---------------------------------------------------------------------------
// fp32 WMMA: D(16x16) = A(16x4) * B(4x16) + C, wave32, ISA 7.12.2 layouts.
// ---------------------------------------------------------------------------
__device__ __forceinline__ v8f wmma_f32(v2f a, v2f b, v8f c) {
  return __builtin_amdgcn_wmma_f32_16x16x4_f32(
      /*neg_a=*/false, a, /*neg_b=*/false, b,
      /*c_mod=*/(short)0, c, /*reuse_a=*/false, /*reuse_b=*/false);
}

// ---------------------------------------------------------------------------
// Kernel 1: fused QKV projection.  y = x @ W.T + b.
// One wave computes a 16x64 strip (4 WMMA accumulators) -> A-frag reused 4x.
// Strips are head-aligned (64 cols) so the [b,h,t,d] scatter is simple.
// ---------------------------------------------------------------------------
__global__ void __launch_bounds__(256)
qkv_proj_kernel(const float* __restrict__ x,
                const float* __restrict__ Wq, const float* __restrict__ bq,
                const float* __restrict__ Wk, const float* __restrict__ bk,
                const float* __restrict__ Wv, const float* __restrict__ bv,
                float* __restrict__ Q, float* __restrict__ K,
                float* __restrict__ V) {
  const int lane = threadIdx.x & 31;
  const int wave = threadIdx.x >> 5;
  const int l    = lane & 15;
  const int half = lane >> 4;

  const int stripsPerMat = (BT / 16) * (D_MODEL / D_K);  // 512*16 = 8192
  long long sid = (long long)blockIdx.x * 8 + wave;      // 3*8192 waves
  const int mat = (int)(sid / stripsPerMat);
  const int rem = (int)(sid % stripsPerMat);
  const int rowBase = (rem / N_HEADS) * 16;
  const int strip   = rem % N_HEADS;                     // head index

  const float* W    = (mat == 0) ? Wq : (mat == 1) ? Wk : Wv;
  const float* bias = (mat == 0) ? bq : (mat == 1) ? bk : bv;
  float*       dst  = (mat == 0) ? Q  : (mat == 1) ? K  : V;

  const float* arow = x + (long long)(rowBase + l) * D_MODEL;
  const float* brow[4];
#pragma unroll
  for (int j = 0; j < 4; ++j)
    brow[j] = W + (long long)(strip * D_K + 16 * j + l) * D_MODEL;

  v8f acc[4];
#pragma unroll
  for (int j = 0; j < 4; ++j) acc[j] = (v8f){0.f,0.f,0.f,0.f,0.f,0.f,0.f,0.f};

  for (int k0 = 0; k0 < D_MODEL; k0 += 4) {
    const int kA = k0 + 2 * half;
    v2f a = *(const v2f*)(arow + kA);
    __builtin_prefetch(arow + kA + 256, 0, 1);   // -> global_prefetch_b8
#pragma unroll
    for (int j = 0; j < 4; ++j) {
      v2f bf = *(const v2f*)(brow[j] + kA);
      acc[j] = wmma_f32(a, bf, acc[j]);
    }
  }

#pragma unroll
  for (int j = 0; j < 4; ++j) {
    const float bn = bias[strip * D_K + 16 * j + l];
#pragma unroll
    for (int r = 0; r < 8; ++r) {
      const int m  = rowBase + r + 8 * half;
      const int bi = m / SEQ;
      const int t  = m % SEQ;
      dst[(((long long)bi * N_HEADS + strip) * SEQ + t) * D_K + 16 * j + l] =
          acc[j][r] + bn;
    }
  }
}

// ---------------------------------------------------------------------------
// Kernel 2: FUSED attention: scores -> softmax -> attn write -> P @ V.
// One block owns 16 query rows of one (b,h).  The 16x2048 probability stripe
// lives in LDS (16*2052*4 B = 128.25 KB; CDNA5 WGP has 320 KB), so the attn
// tensor hits HBM exactly once (the mandatory output write).
// ---------------------------------------------------------------------------
__global__ void __launch_bounds__(256)
attn_fused_kernel(const float* __restrict__ Q, const float* __restrict__ K,
                  const float* __restrict__ V, float* __restrict__ attn,
                  float* __restrict__ ctx) {
  __shared__ float S[16 * SPAD];            // softmax stripe
  __shared__ float red[2 * 4 * 16 * 16];    // context partial sums (8 KB)

  const int tid  = threadIdx.x;
  const int lane = tid & 31;
  const int wave = tid >> 5;
  const int l    = lane & 15;
  const int half = lane >> 4;

  const int bh      = blockIdx.x >> 7;           // 128 row tiles per (b,h)
  const int rowBase = (blockIdx.x & 127) * 16;
  const int b = bh / N_HEADS, h = bh % N_HEADS;

  const float* Qh = Q + (long long)bh * SEQ * D_K;
  const float* Kh = K + (long long)bh * SEQ * D_K;
  const float* Vh = V + (long long)bh * SEQ * D_K;

  // ---- Phase 1: S = scale * (Q_tile @ K^T) ------------------------------
  // Whole 16x64 Q tile hoisted into 16 A-fragments (32 VGPRs), reused for
  // all 16 column tiles this wave produces.
  v2f aQ[16];
  {
    const float* arow = Qh + (long long)(rowBase + l) * D_K;
#pragma unroll
    for (int kk = 0; kk < 16; ++kk)
      aQ[kk] = *(const v2f*)(arow + 4 * kk + 2 * half);
  }
  for (int i = 0; i < 16; ++i) {                 // 8 waves * 16 tiles = 2048 cols
    const int ct = wave * 16 + i;
    const float* brow = Kh + (long long)(ct * 16 + l) * D_K;
    v8f acc = {0.f,0.f,0.f,0.f,0.f,0.f,0.f,0.f};
#pragma unroll
    for (int kk = 0; kk < 16; ++kk) {
      v2f bf = *(const v2f*)(brow + 4 * kk + 2 * half);
      acc = wmma_f32(aQ[kk], bf, acc);
    }
#pragma unroll
    for (int r = 0; r < 8; ++r)
      S[(r + 8 * half) * SPAD + ct * 16 + l] = acc[r] * 0.125f;  // 1/sqrt(64)
  }
  __syncthreads();

  // ---- Phase 2: row softmax (wave32 shuffle reductions, 2 rows/wave) ----
#pragma unroll
  for (int rr = 0; rr < 2; ++rr) {
    const int row = wave * 2 + rr;
    float* p = &S[row * SPAD];
    float mx = -3.402823466e38f;
    for (int c = lane; c < SEQ; c += 32) mx = fmaxf(mx, p[c]);
#pragma unroll
    for (int o = 16; o > 0; o >>= 1) mx = fmaxf(mx, __shfl_xor(mx, o, 32));
    float sum = 0.f;
    for (int c = lane; c < SEQ; c += 32) {
      const float e = __expf(p[c] - mx);
      p[c] = e;
      sum += e;
    }
#pragma unroll
    for (int o = 16; o > 0; o >>= 1) sum += __shfl_xor(sum, o, 32);
    const float inv = 1.0f / sum;
    for (int c = lane; c < SEQ; c += 32) p[c] *= inv;
  }
  __syncthreads();

  // ---- Phase 3: single write of attn weights to d_out -------------------
  {
    float* dst = attn + ((long long)bh * SEQ + rowBase) * SEQ;  // 16 rows contig
    for (int e = tid; e < 16 * SEQ; e += 256) {
      const int row = e >> 11;          // e / 2048
      const int col = e & (SEQ - 1);
      dst[e] = S[row * SPAD + col];
    }
  }

  // ---- Phase 4: context = P @ V straight out of LDS ---------------------
  // wave -> (khalf = wave/4 in {0,1}, ct = wave%4): K-split by 2, 4 col tiles.
  {
    const int ct    = wave & 3;
    const int khalf = wave >> 2;
    v8f acc = {0.f,0.f,0.f,0.f,0.f,0.f,0.f,0.f};
    const int kBeg = khalf * (SEQ / 2);
    for (int k0 = kBeg; k0 < kBeg + SEQ / 2; k0 += 4) {
      const int kA = k0 + 2 * half;
      v2f a = *(const v2f*)(&S[l * SPAD + kA]);              // ds_load_b64
      v2f bf;
      bf.x = Vh[(long long)kA * D_K + ct * 16 + l];
      bf.y = Vh[(long long)(kA + 1) * D_K + ct * 16 + l];
      acc = wmma_f32(a, bf, acc);
    }
    float* my = &red[wave * 256];
#pragma unroll
    for (int r = 0; r < 8; ++r) my[(r + 8 * half) * 16 + l] = acc[r];
  }
  __syncthreads();

  // combine the two K-halves and write ctx[b, t, h*64 + d]
  for (int e = tid; e < 1024; e += 256) {
    const float vsum = red[e] + red[e + 1024];
    const int ctIdx = e >> 8;
    const int row   = (e >> 4) & 15;
    const int col   = e & 15;
    ctx[((long long)b * SEQ + rowBase + row) * D_MODEL +
        h * D_K + ctIdx * 16 + col] = vsum;
  }
}

// ---------------------------------------------------------------------------
// Kernel 3: out = ctx @ Wo.T + bo.  Same 16x64-strip register blocking.
// ---------------------------------------------------------------------------
__global__ void __launch_bounds__(256)
outproj_kernel(const float* __restrict__ ctx, const float* __restrict__ Wo,
               const float* __restrict__ bo, float* __restrict__ out) {
  const int lane = threadIdx.x & 31;
  const int wave = threadIdx.x >> 5;
  const int l    = lane & 15;
  const int half = lane >> 4;

  long long sid = (long long)blockIdx.x * 8 + wave;   // 512*16 = 8192 strips
  const int rowBase = (int)(sid / 16) * 16;
  const int strip   = (int)(sid % 16);

  const float* arow = ctx + (long long)(rowBase + l) * D_MODEL;
  const float* brow[4];
#pragma unroll
  for (int j = 0; j < 4; ++j)
    brow[j] = Wo + (long long)(strip * 64 + 16 * j + l) * D_MODEL;

  v8f acc[4];
#pragma unroll
  for (int j = 0; j < 4; ++j) acc[j] = (v8f){0.f,0.f,0.f,0.f,0.f,0.f,0.f,0.f};

  for (int k0 = 0; k0 < D_MODEL; k0 += 4) {
    const int kA = k0 + 2 * half;
    v2f a = *(const v2f*)(arow + kA);
    __builtin_prefetch(arow + kA + 256, 0, 1);
#pragma unroll
    for (int j = 0; j < 4; ++j) {
      v2f bf = *(const v2f*)(brow[j] + kA);
      acc[j] = wmma_f32(a, bf, acc[j]);
    }
  }

#pragma unroll
  for (int j = 0; j < 4; ++j) {
    const float bn = bo[strip * 64 + 16 * j + l];
#pragma unroll
    for (int r = 0; r < 8; ++r) {
      const int m = rowBase + r + 8 * half;
      out[(long long)m * D_MODEL + strip * 64 + 16 * j + l] = acc[j][r] + bn;
    }
  }
}

// ---------------------------------------------------------------------------
// Kernel 4: y = LayerNorm(out + x) * g + b ; one block per token row.
// ---------------------------------------------------------------------------
__global__ void __launch_bounds__(256)
add_ln_kernel(const float* __restrict__ out, const float* __restrict__ x,
              const float* __restrict__ g, const float* __restrict__ bta,
              float* __restrict__ y) {
  __shared__ float red1[256];
  __shared__ float red2[256];
  const int tid = threadIdx.x;
  const long long row = blockIdx.x;
  const float* o  = out + row * D_MODEL;
  const float* xi = x   + row * D_MODEL;

  float v[4];
  float s = 0.f, s2 = 0.f;
#pragma unroll
  for (int i = 0; i < 4; ++i) {
    const int c = tid + i * 256;
    v[i] = o[c] + xi[c];
    s  += v[i];
    s2 += v[i] * v[i];
  }
  red1[tid] = s;
  red2[tid] = s2;
  __syncthreads();
  for (int st = 128; st > 0; st >>= 1) {
    if (tid < st) {
      red1[tid] += red1[tid + st];
      red2[tid] += red2[tid + st];
    }
    __syncthreads();
  }
  const float mu  = red1[0] * (1.0f / D_MODEL);
  const float var = red2[0] * (1.0f / D_MODEL) - mu * mu;
  const float inv = rsqrtf(var + LN_EPS);
#pragma unroll
  for (int i = 0; i < 4; ++i) {
    const int c = tid + i * 256;
    y[row * D_MODEL + c] = (v[i] - mu) * inv * g[c] + bta[c];
  }
}

// ---------------------------------------------------------------------------
// Launch
// ---------------------------------------------------------------------------
extern "C" void kernel_launch(void* const* d_in, const int* in_sizes, int n_in,
                              void* d_out, int out_size, void* d_ws, size_t ws_size,
                              hipStream_t stream) {
  (void)in_sizes; (void)n_in; (void)out_size; (void)ws_size;

  const float* x    = (const float*)d_in[0];
  const float* Wq   = (const float*)d_in[1];
  const float* bq   = (const float*)d_in[2];
  const float* Wk   = (const float*)d_in[3];
  const float* bk   = (const float*)d_in[4];
  const float* Wv   = (const float*)d_in[5];
  const float* bv   = (const float*)d_in[6];
  const float* Wo   = (const float*)d_in[7];
  const float* bo   = (const float*)d_in[8];
  const float* ln_g = (const float*)d_in[9];
  const float* ln_b = (const float*)d_in[10];

  float* y    = (float*)d_out;                       // [4,2048,1024]
  float* attn = y + (long long)BT * D_MODEL;         // [4,16,2048,2048]

  const long long MATSZ = (long long)BT * D_MODEL;   // 8,388,608 floats
  float* ws  = (float*)d_ws;
  float* Q   = ws;                 // [b,h,t,d]
  float* K   = ws + MATSZ;
  float* V   = ws + 2 * MATSZ;
  float* ctx = ws + 3 * MATSZ;     // Q is live during fused kernel
  float* out = ws;                 // Q dead after fused kernel -> reuse

  const dim3 blk(256);

  // 1) QKV projections: 3*8192 strips / 8 waves = 3072 blocks
  qkv_proj_kernel<<<dim3(3072), blk, 0, stream>>>(x, Wq, bq, Wk, bk, Wv, bv, Q, K, V);

  // 2) fused scores+softmax+attn-write+context: 64 bh * 128 row tiles
  attn_fused_kernel<<<dim3(8192), blk, 0, stream>>>(Q, K, V, attn, ctx);

  // 3) output projection: 8192 strips / 8 = 1024 blocks
  outproj_kernel<<<dim3(1024), blk, 0, stream>>>(ctx, Wo, bo, out);

  // 4) residual + LayerNorm: one block per token (8192)
  add_ln_kernel<<<dim3(8192), blk, 0, stream>>>(out, x, ln_g, ln_b, y);
}